// VGAEEncoder_31344671326932
// MI455X (gfx1250) — compile-verified
//
#include <hip/hip_runtime.h>

// VGAE encoder for MI455X (gfx1250, wave32).
// - GEMMs: V_WMMA_F32_16X16X4_F32 (f32 in/out), one wave per 16x128 output strip.
// - W (64 KB) staged into LDS via the Tensor Data Mover (tensor_load_to_lds),
//   tracked with TENSORcnt; falls back to manual copy if the builtin is absent.
// - Edge aggregation: L2-resident scatter-add with global_atomic_add_f32.

typedef __attribute__((ext_vector_type(2))) float v2f;
typedef __attribute__((ext_vector_type(8))) float v8f;
typedef __attribute__((ext_vector_type(4))) unsigned int u32x4;
typedef __attribute__((ext_vector_type(4))) int i32x4;
typedef __attribute__((ext_vector_type(8))) int i32x8;

#define D 128  // feature dim (D_IN == D_OUT == 128)

// ---------------- small elementwise kernels ----------------

__global__ void fill_kernel(float* __restrict__ p, float v, int n) {
  int i = blockIdx.x * blockDim.x + threadIdx.x;
  if (i < n) p[i] = v;
}

__global__ void degree_kernel(const int* __restrict__ dst, float* __restrict__ deg, int E) {
  int i = blockIdx.x * blockDim.x + threadIdx.x;
  if (i < E) atomicAdd(&deg[dst[i]], 1.0f);
}

__global__ void rsqrt_kernel(const float* __restrict__ deg, float* __restrict__ dinv, int n) {
  int i = blockIdx.x * blockDim.x + threadIdx.x;
  if (i < n) dinv[i] = rsqrtf(deg[i]);
}

// out[n][c] = t[n][c] * dinv[n]^2 + bias[c]   (self-loop term + bias, scatter accumulates on top)
__global__ void preinit_kernel(const float* __restrict__ t, const float* __restrict__ dinv,
                               const float* __restrict__ bias, float* __restrict__ out, int total) {
  int i = blockIdx.x * blockDim.x + threadIdx.x;
  if (i < total) {
    int n = i >> 7;
    int c = i & (D - 1);
    float dv = dinv[n];
    out[i] = t[i] * dv * dv + bias[c];
  }
}

__global__ void relu_kernel(float* __restrict__ p, int total) {
  int i = blockIdx.x * blockDim.x + threadIdx.x;
  if (i < total) p[i] = fmaxf(p[i], 0.0f);
}

// ---------------- WMMA f32 GEMM: H[nrows x 128] = X[nrows x 128] @ W[128 x 128] ----------------

__global__ __launch_bounds__(256) void gemm_wmma_kernel(const float* __restrict__ X,
                                                        const float* __restrict__ W,
                                                        float* __restrict__ H, int nrows) {
  __shared__ float wlds[D * D];  // 64 KB
  const int tid = threadIdx.x;
  const int wave = tid >> 5;
  const int lane = tid & 31;

#if __has_builtin(__builtin_amdgcn_tensor_load_to_lds) && \
    __has_builtin(__builtin_amdgcn_s_wait_tensorcnt)
  // Stage W via the Tensor Data Mover: 2D tile 128x128, f32, row stride 128.
  if (wave == 0) {
    const unsigned lds_base = (unsigned)(uintptr_t)(&wlds[0]);        // LDS byte address
    const unsigned long long ga = (unsigned long long)(uintptr_t)W;   // global byte address
    // D# group0: count=1 | lds_addr | global_addr[56:0] | type=2 (bits 127:126)
    u32x4 g0 = { 1u,
                 lds_base,
                 (unsigned)(ga & 0xFFFFFFFFu),
                 (unsigned)((ga >> 32) & 0x01FFFFFFu) | 0x80000000u };
    // D# group1: data_size=2(4B)<<16 | tensor_dim0=128<<16 | tensor_dim1=128<<16 |
    //            tile_dim0=128<<16 | tile_dim1=128 | tensor_dim0_stride=128
    i32x8 g1 = { (int)0x00020000,
                 (int)(128u << 16),
                 (int)(128u << 16),
                 (int)(128u << 16),
                 (int)128,
                 (int)128,
                 0, 0 };
    i32x4 z4 = { 0, 0, 0, 0 };
#if defined(__clang_major__) && (__clang_major__ >= 23)
    i32x8 z8 = { 0, 0, 0, 0, 0, 0, 0, 0 };
    __builtin_amdgcn_tensor_load_to_lds(g0, g1, z4, z4, z8, 0);
#else
    __builtin_amdgcn_tensor_load_to_lds(g0, g1, z4, z4, 0);
#endif
    __builtin_amdgcn_s_wait_tensorcnt(0);
  }
  __syncthreads();
#else
  for (int i = tid; i < D * D; i += 256) wlds[i] = W[i];
  __syncthreads();
#endif

  const int row0 = (blockIdx.x * 8 + wave) * 16;
  if (row0 >= nrows) return;  // uniform per-wave: EXEC stays all-ones for WMMA

  const int m = lane & 15;            // A row within tile / B-D column within tile
  const int khalf = (lane >> 4) << 1; // lanes 0-15 -> K 0,1 ; lanes 16-31 -> K 2,3
  int arow = row0 + m;
  if (arow >= nrows) arow = nrows - 1;  // clamp (50000 % 16 == 0, defensive only)
  const float* __restrict__ xrow = X + (size_t)arow * D;

  v8f acc[8];
  v8f zero = 0;
#pragma unroll
  for (int t = 0; t < 8; ++t) acc[t] = zero;

  for (int k0 = 0; k0 < D; k0 += 4) {
    v2f A;
    A.x = xrow[k0 + khalf];
    A.y = xrow[k0 + khalf + 1];
#pragma unroll
    for (int t = 0; t < 8; ++t) {
      v2f B;
      B.x = wlds[(k0 + khalf) * D + t * 16 + m];
      B.y = wlds[(k0 + khalf + 1) * D + t * 16 + m];
      // (neg_a, A, neg_b, B, c_mod, C, reuse_a, reuse_b)
      acc[t] = __builtin_amdgcn_wmma_f32_16x16x4_f32(false, A, false, B, (short)0, acc[t],
                                                     false, false);
    }
  }

  // D tile layout: VGPR i -> lanes 0-15: (M=i, N=lane); lanes 16-31: (M=i+8, N=lane-16)
  const int rbase = row0 + ((lane >> 4) << 3);
#pragma unroll
  for (int t = 0; t < 8; ++t) {
    const int c = t * 16 + m;
#pragma unroll
    for (int i = 0; i < 8; ++i) {
      const int r = rbase + i;
      if (r < nrows) H[(size_t)r * D + c] = acc[t][i];
    }
  }
}

// ---------------- edge scatter: out[dst] += h[src] * dinv[src]*dinv[dst] ----------------
// One wave per edge; lane handles 4 channels (32 lanes * 4 = 128).

__global__ __launch_bounds__(256) void scatter_kernel(const float* __restrict__ h,
                                                      const int* __restrict__ src,
                                                      const int* __restrict__ dst,
                                                      const float* __restrict__ dinv,
                                                      float* __restrict__ out, int E) {
  const int lane = threadIdx.x & 31;
  const int warp = (blockIdx.x * blockDim.x + threadIdx.x) >> 5;
  const int nwarp = (gridDim.x * blockDim.x) >> 5;
  for (int e = warp; e < E; e += nwarp) {
    const int s = src[e];
    const int d = dst[e];
    const float norm = dinv[s] * dinv[d];
    const float4 v = ((const float4*)(h + (size_t)s * D))[lane];
    float* op = out + (size_t)d * D + lane * 4;
    atomicAdd(op + 0, v.x * norm);
    atomicAdd(op + 1, v.y * norm);
    atomicAdd(op + 2, v.z * norm);
    atomicAdd(op + 3, v.w * norm);
  }
}

// ---------------- driver ----------------

extern "C" void kernel_launch(void* const* d_in, const int* in_sizes, int n_in,
                              void* d_out, int out_size, void* d_ws, size_t ws_size,
                              hipStream_t stream) {
  const float* x   = (const float*)d_in[0];
  const int*   ei  = (const int*)d_in[1];
  const float* W1  = (const float*)d_in[2];
  const float* b1  = (const float*)d_in[3];
  const float* Wmu = (const float*)d_in[4];
  const float* bmu = (const float*)d_in[5];
  const float* Wls = (const float*)d_in[6];
  const float* bls = (const float*)d_in[7];

  const int N = in_sizes[0] / D;
  const int E = in_sizes[1] / 2;
  const int* src = ei;       // edge_index[0]
  const int* dst = ei + E;   // edge_index[1]

  float* wsf  = (float*)d_ws;
  float* dinv = wsf;                      // N floats
  float* hbuf = wsf + (size_t)N;          // N*D floats (hidden, post-ReLU)
  float* tmp  = hbuf + (size_t)N * D;     // N*D floats (GEMM output)
  float* deg  = tmp;                      // reuse tmp region before first GEMM

  float* mu = (float*)d_out;
  float* ls = (float*)d_out + (size_t)N * D;

  const int total = N * D;
  const dim3 b256(256);
  const int gElem  = (total + 255) / 256;
  const int gNode  = (N + 255) / 256;
  const int gEdge  = (E + 255) / 256;
  const int gGemm  = ((N + 15) / 16 + 7) / 8;   // 8 waves (16-row tiles) per block
  const int gScat  = (E + 7) / 8;               // 8 warps per 256-thread block, 1 edge each

  // normalization: deg = 1 + in-degree ; dinv = rsqrt(deg)
  fill_kernel<<<gNode, b256, 0, stream>>>(deg, 1.0f, N);
  degree_kernel<<<gEdge, b256, 0, stream>>>(dst, deg, E);
  rsqrt_kernel<<<gNode, b256, 0, stream>>>(deg, dinv, N);

  // conv1: hbuf = relu( scatter(x@W1) + (x@W1)*dinv^2 + b1 )
  gemm_wmma_kernel<<<gGemm, b256, 0, stream>>>(x, W1, tmp, N);
  preinit_kernel<<<gElem, b256, 0, stream>>>(tmp, dinv, b1, hbuf, total);
  scatter_kernel<<<gScat, b256, 0, stream>>>(tmp, src, dst, dinv, hbuf, E);
  relu_kernel<<<gElem, b256, 0, stream>>>(hbuf, total);

  // conv mu -> d_out[0 : N*D)
  gemm_wmma_kernel<<<gGemm, b256, 0, stream>>>(hbuf, Wmu, tmp, N);
  preinit_kernel<<<gElem, b256, 0, stream>>>(tmp, dinv, bmu, mu, total);
  scatter_kernel<<<gScat, b256, 0, stream>>>(tmp, src, dst, dinv, mu, E);

  // conv logstd -> d_out[N*D : 2*N*D)
  gemm_wmma_kernel<<<gGemm, b256, 0, stream>>>(hbuf, Wls, tmp, N);
  preinit_kernel<<<gElem, b256, 0, stream>>>(tmp, dinv, bls, ls, total);
  scatter_kernel<<<gScat, b256, 0, stream>>>(tmp, src, dst, dinv, ls, E);
}